// MusicTheoryLoss_59854664237399
// MI455X (gfx1250) — compile-verified
//
#include <hip/hip_runtime.h>

typedef float v2f __attribute__((ext_vector_type(2)));
typedef float v8f __attribute__((ext_vector_type(8)));

#define MTL_BLOCKS  512
#define MTL_THREADS 256
#define MTL_WPB     (MTL_THREADS / 32)

// interval table {1.0,0.5,0.3,0.2,0.1,0.1,0.7,0.0,0.2,0.3,0.4,0.5} packed as
// nibbles (value*10); 0.1f * n reproduces each f32 table entry exactly.
#define MTL_PACK 0x54320711235AULL

__global__ __launch_bounds__(MTL_THREADS) void mtl_partial_kernel(
    const float* __restrict__ outputs,
    const float* __restrict__ targets,
    const float* __restrict__ melodies,
    float2* __restrict__ partials,
    int R)
{
    const int tid  = blockIdx.x * MTL_THREADS + threadIdx.x;
    const int nthr = gridDim.x * MTL_THREADS;
    const int lane = threadIdx.x & 31;
    const int wid  = threadIdx.x >> 5;

    // 4 independent MSE accumulators -> packable FMA chains
    float ms0 = 0.0f, ms1 = 0.0f, ms2 = 0.0f, ms3 = 0.0f;
    float harm_acc = 0.0f;

    for (int r = tid; r < R; r += nthr) {
        const float4* __restrict__ orow = (const float4*)(outputs  + (size_t)r * 128);
        const float4* __restrict__ trow = (const float4*)(targets  + (size_t)r * 128);
        const float4* __restrict__ mrow = (const float4*)(melodies + (size_t)r * 128);

        // speculative prefetch of this thread's next row (silently dropped if OOB)
        {
            const size_t nb = (size_t)(r + nthr) * 128;
            __builtin_prefetch(outputs  + nb, 0, 0);
            __builtin_prefetch(targets  + nb, 0, 0);
            __builtin_prefetch(melodies + nb, 0, 0);
        }

        float hbv = -__builtin_inff(); int hbi = 0;   // harmony (outputs) argmax
        float mbv = -__builtin_inff(); int mbi = 0;   // melody argmax

        #pragma unroll
        for (int k = 0; k < 32; ++k) {
            const float4 o = orow[k];
            const float4 t = trow[k];
            const float4 m = mrow[k];

            const float d0 = o.x - t.x, d1 = o.y - t.y, d2 = o.z - t.z, d3 = o.w - t.w;
            ms0 = fmaf(d0, d0, ms0);
            ms1 = fmaf(d1, d1, ms1);
            ms2 = fmaf(d2, d2, ms2);
            ms3 = fmaf(d3, d3, ms3);

            const int b = k * 4;
            // strict > keeps the FIRST max index, matching jnp.argmax
            if (o.x > hbv) { hbv = o.x; hbi = b + 0; }
            if (o.y > hbv) { hbv = o.y; hbi = b + 1; }
            if (o.z > hbv) { hbv = o.z; hbi = b + 2; }
            if (o.w > hbv) { hbv = o.w; hbi = b + 3; }

            if (m.x > mbv) { mbv = m.x; mbi = b + 0; }
            if (m.y > mbv) { mbv = m.y; mbi = b + 1; }
            if (m.z > mbv) { mbv = m.z; mbi = b + 2; }
            if (m.w > mbv) { mbv = m.w; mbi = b + 3; }
        }

        int dlt = hbi - mbi;
        dlt = dlt < 0 ? -dlt : dlt;
        const int interval = dlt % 12;
        harm_acc += 0.1f * (float)((MTL_PACK >> (interval * 4)) & 0xF);
    }

    const float mse_acc = (ms0 + ms1) + (ms2 + ms3);

    // ---- cross-lane reduction of both partials via V_WMMA_F32_16X16X4_F32 ----
    // A (16x4 f32): lanes 0-15 hold (K0,K1) of row M=lane; lanes 16-31 hold
    // (K2,K3) of row M=lane-16. With B = all-ones (layout-invariant) every D
    // column holds D[m][*] = acc[m] + acc[m+16]; summing the 8 D VGPRs plus one
    // half-wave exchange yields the full 32-lane sum in fp32.
    v2f ones; ones[0] = 1.0f; ones[1] = 1.0f;

    v2f a; a[0] = mse_acc; a[1] = 0.0f;
    v8f c = {};
    c = __builtin_amdgcn_wmma_f32_16x16x4_f32(false, a, false, ones,
                                              (short)0, c, false, false);
    float msew = ((c[0] + c[1]) + (c[2] + c[3])) + ((c[4] + c[5]) + (c[6] + c[7]));
    msew += __shfl_xor(msew, 16, 32);

    v2f ah; ah[0] = harm_acc; ah[1] = 0.0f;
    v8f ch = {};
    ch = __builtin_amdgcn_wmma_f32_16x16x4_f32(false, ah, false, ones,
                                               (short)0, ch, false, false);
    float harmw = ((ch[0] + ch[1]) + (ch[2] + ch[3])) + ((ch[4] + ch[5]) + (ch[6] + ch[7]));
    harmw += __shfl_xor(harmw, 16, 32);

    // ---- block reduction, deterministic order ----
    __shared__ float2 sp[MTL_WPB];
    if (lane == 0) sp[wid] = make_float2(msew, harmw);
    __syncthreads();
    if (threadIdx.x == 0) {
        float sa = 0.0f, sb = 0.0f;
        #pragma unroll
        for (int i = 0; i < MTL_WPB; ++i) { sa += sp[i].x; sb += sp[i].y; }
        partials[blockIdx.x] = make_float2(sa, sb);
    }
}

__global__ __launch_bounds__(256) void mtl_final_kernel(
    const float2* __restrict__ partials,
    float* __restrict__ out,
    int G, float invN, float invR)
{
    __shared__ float2 s[256];
    float sa = 0.0f, sb = 0.0f;
    for (int i = threadIdx.x; i < G; i += 256) {
        sa += partials[i].x;
        sb += partials[i].y;
    }
    s[threadIdx.x] = make_float2(sa, sb);
    __syncthreads();
    #pragma unroll
    for (int off = 128; off > 0; off >>= 1) {
        if ((int)threadIdx.x < off) {
            s[threadIdx.x].x += s[threadIdx.x + off].x;
            s[threadIdx.x].y += s[threadIdx.x + off].y;
        }
        __syncthreads();
    }
    if (threadIdx.x == 0) {
        // ALPHA * mse_mean + BETA * harmony_mean (ALPHA = BETA = 1)
        out[0] = s[0].x * invN + s[0].y * invR;
    }
}

extern "C" void kernel_launch(void* const* d_in, const int* in_sizes, int n_in,
                              void* d_out, int out_size, void* d_ws, size_t ws_size,
                              hipStream_t stream) {
    const float* outputs  = (const float*)d_in[0];
    const float* targets  = (const float*)d_in[1];
    const float* melodies = (const float*)d_in[2];
    const int N = in_sizes[0];   // B*1*S*P = 16,777,216
    const int R = N / 128;       // B*S rows = 131,072

    float2* partials = (float2*)d_ws;

    mtl_partial_kernel<<<MTL_BLOCKS, MTL_THREADS, 0, stream>>>(
        outputs, targets, melodies, partials, R);
    mtl_final_kernel<<<1, 256, 0, stream>>>(
        partials, (float*)d_out, MTL_BLOCKS, 1.0f / (float)N, 1.0f / (float)R);
}